// DeepGPClassificationModel_23699629539572
// MI455X (gfx1250) — compile-verified
//
#include <hip/hip_runtime.h>
#include <math.h>

#define N       2048
#define DIN     512
#define MF      128       // feature dim m
#define NB      64        // GJ block size
#define NBLK    (N / NB)  // 32
#define NSTEPS  50

typedef float v2f __attribute__((ext_vector_type(2)));
typedef float v8f __attribute__((ext_vector_type(8)));

// ---- CDNA5 async global->LDS path (guarded; ds-store fallback otherwise)
#if defined(__has_builtin)
#if __has_builtin(__builtin_amdgcn_global_load_async_to_lds_b32)
#define ASYNC_LDS 1
#endif
#endif
#ifndef ASYNC_LDS
#define ASYNC_LDS 0
#endif

// f32 WMMA: D(16x16,f32) = A(16x4,f32) * B(4x16,f32) + C
// A-frag: lane holds A[M = lane&15, K = 2*(lane>>4) + {0,1}]  (ISA 7.12.2)
// B-frag: lane holds B[K = 2*(lane>>4) + {0,1}, N = lane&15]
// C/D   : vgpr r -> D[M = r + 8*(lane>>4), N = lane&15]
__device__ __forceinline__ v8f wmma4(v2f a, v2f b, v8f c) {
  return __builtin_amdgcn_wmma_f32_16x16x4_f32(false, a, false, b, (short)0, c,
                                               false, false);
}

// Stage one 64x64 f32 tile from global (row stride gstride) into LDS with
// padded stride NB+1 (padding removes the 16-way bank conflict on A-frag
// column reads). Async path: bytes go straight to LDS, tracked by ASYNCcnt.
__device__ __forceinline__ void tile_to_lds(float* __restrict__ sdst,
                                            const float* __restrict__ gsrc,
                                            int gstride) {
#if ASYNC_LDS
  for (int idx = threadIdx.x; idx < NB * NB; idx += 256) {
    int r = idx >> 6, c = idx & 63;
    __builtin_amdgcn_global_load_async_to_lds_b32(
        (__attribute__((address_space(1))) int*)(uintptr_t)(
            gsrc + (size_t)r * gstride + c),
        (__attribute__((address_space(3))) int*)(uint32_t)(uintptr_t)(
            sdst + r * (NB + 1) + c),
        0, 0);
  }
#else
  for (int idx = threadIdx.x; idx < NB * NB; idx += 256) {
    int r = idx >> 6, c = idx & 63;
    sdst[r * (NB + 1) + c] = gsrc[(size_t)r * gstride + c];
  }
#endif
}

// Wave waits for its own async transfers, then block-wide barrier covers all.
__device__ __forceinline__ void tile_sync() {
#if ASYNC_LDS
#if __has_builtin(__builtin_amdgcn_s_wait_asynccnt)
  __builtin_amdgcn_s_wait_asynccnt(0);
#else
  asm volatile("s_wait_asynccnt 0x0" ::: "memory");
#endif
#endif
  __syncthreads();
}

// ---------------------------------------------------------------- utilities
__global__ void zero4(float4* __restrict__ dst) {
  size_t i = (size_t)blockIdx.x * 256 + threadIdx.x;
  dst[i] = make_float4(0.f, 0.f, 0.f, 0.f);
}
__global__ void copy_mat4(const float4* __restrict__ s, float4* __restrict__ d) {
  size_t i = (size_t)blockIdx.x * 256 + threadIdx.x;
  d[i] = s[i];
}
__global__ void copy_vec(const float* __restrict__ s, float* __restrict__ d) {
  int i = blockIdx.x * 256 + threadIdx.x;
  if (i < N) d[i] = s[i];
}

// ------------------------------------------------- proj = x @ W_fe + b_fe
__global__ __launch_bounds__(256) void gemm_xw(const float* __restrict__ x,
                                               const float* __restrict__ W,
                                               const float* __restrict__ bias,
                                               float* __restrict__ P) {
  int wave = threadIdx.x >> 5, lane = threadIdx.x & 31;
  int tid = blockIdx.x * 8 + wave;            // 1024 16x16 tiles
  int tM = (tid >> 3) << 4;                   // 128 row tiles
  int tN = (tid & 7) << 4;                    // 8 col tiles
  int l15 = lane & 15, kh = (lane >> 4) << 1;
  const float* arow = x + (size_t)(tM + l15) * DIN;
  v8f acc = {};
  for (int k = 0; k < DIN; k += 4) {
    v2f a, b;
    a.x = arow[k + kh];
    a.y = arow[k + kh + 1];
    b.x = W[(size_t)(k + kh) * MF + tN + l15];
    b.y = W[(size_t)(k + kh + 1) * MF + tN + l15];
    acc = wmma4(a, b, acc);
  }
  int rbase = tM + ((lane >> 4) << 3), col = tN + l15;
  float bv = bias[col];
#pragma unroll
  for (int r = 0; r < 8; ++r) P[(size_t)(rbase + r) * MF + col] = acc[r] + bv;
}

// ------------------------------------------------- per-column min / max
__global__ __launch_bounds__(256) void col_minmax(const float* __restrict__ P,
                                                  float* __restrict__ cmin,
                                                  float* __restrict__ cmax) {
  __shared__ float smn[256], smx[256];
  int c = blockIdx.x;
  float mn = 3.4e38f, mx = -3.4e38f;
  for (int r = threadIdx.x; r < N; r += 256) {
    float v = P[(size_t)r * MF + c];
    mn = fminf(mn, v);
    mx = fmaxf(mx, v);
  }
  smn[threadIdx.x] = mn; smx[threadIdx.x] = mx;
  __syncthreads();
  for (int s = 128; s > 0; s >>= 1) {
    if (threadIdx.x < s) {
      smn[threadIdx.x] = fminf(smn[threadIdx.x], smn[threadIdx.x + s]);
      smx[threadIdx.x] = fmaxf(smx[threadIdx.x], smx[threadIdx.x + s]);
    }
    __syncthreads();
  }
  if (threadIdx.x == 0) { cmin[c] = smn[0]; cmax[c] = smx[0]; }
}

// ------------------- normalize to [-1,1]*sqrt(kvar); emit G = projT output
__global__ void normalize_G(float* __restrict__ P, const float* __restrict__ cmin,
                            const float* __restrict__ cmax,
                            const float* __restrict__ kvar,
                            float* __restrict__ Gout) {
  int idx = blockIdx.x * 256 + threadIdx.x;   // N*MF
  int r = idx / MF, c = idx % MF;
  float mn = cmin[c], mx = cmax[c];
  float v = (2.0f * (P[idx] - mn) / (mx - mn) - 1.0f) * sqrtf(kvar[0]);
  P[idx] = v;
  Gout[(size_t)c * N + r] = v;
}

// ------------------------------------------------- K = P @ P^T + var*I
__global__ __launch_bounds__(256) void syrk_K(const float* __restrict__ P,
                                              const float* __restrict__ var,
                                              float* __restrict__ Mm) {
  int wave = threadIdx.x >> 5, lane = threadIdx.x & 31;
  int tid = blockIdx.x * 8 + wave;            // 128*128 tiles
  int tM = (tid >> 7) << 4, tN = (tid & 127) << 4;
  int l15 = lane & 15, kh = (lane >> 4) << 1;
  const float* arow = P + (size_t)(tM + l15) * MF;
  const float* brow = P + (size_t)(tN + l15) * MF;   // B[k][col] = P[col][k]
  v8f acc = {};
  for (int k = 0; k < MF; k += 4) {
    v2f a, b;
    a.x = arow[k + kh];  a.y = arow[k + kh + 1];
    b.x = brow[k + kh];  b.y = brow[k + kh + 1];
    acc = wmma4(a, b, acc);
  }
  float v0 = var[0];
  int rbase = tM + ((lane >> 4) << 3), col = tN + l15;
#pragma unroll
  for (int r = 0; r < 8; ++r) {
    int row = rbase + r;
    Mm[(size_t)row * N + col] = acc[r] + (row == col ? v0 : 0.0f);
  }
}

// --------------------------- 64x64 pivot-block inverse in LDS (scalar GJ)
__global__ __launch_bounds__(256) void inv_diag(const float* __restrict__ Mm,
                                                int kb, float* __restrict__ binv) {
  __shared__ float s[NB * (NB + 1)];
  __shared__ float fcol[NB];
  __shared__ float pshare;
  const float* blk = Mm + (size_t)kb * NB * N + (size_t)kb * NB;
  tile_to_lds(s, blk, N);
  tile_sync();
  for (int k = 0; k < NB; ++k) {
    if (threadIdx.x == 0) pshare = 1.0f / s[k * (NB + 1) + k];
    __syncthreads();
    float p = pshare;
    if (threadIdx.x < NB) {                       // scale pivot row
      int j = threadIdx.x;
      if (j != k) s[k * (NB + 1) + j] *= p;
    } else if (threadIdx.x < 2 * NB) {            // snapshot pivot column
      int i = threadIdx.x - NB;
      fcol[i] = s[i * (NB + 1) + k];
    }
    __syncthreads();
    int i = threadIdx.x >> 2, jb = (threadIdx.x & 3) << 4;
    if (i != k) {
      float f = fcol[i];
#pragma unroll
      for (int jj = 0; jj < 16; ++jj) {
        int j = jb + jj;
        if (j != k) s[i * (NB + 1) + j] -= f * s[k * (NB + 1) + j];
        else        s[i * (NB + 1) + k] = -f * p;
      }
    }
    __syncthreads();
    if (threadIdx.x == 0) s[k * (NB + 1) + k] = p;
    __syncthreads();
  }
  for (int idx = threadIdx.x; idx < NB * NB; idx += 256) {
    int r = idx >> 6, c = idx & 63;
    binv[idx] = s[r * (NB + 1) + c];
  }
}

// ---------------------- snapshot of pivot column panel (old M[:, kb-block])
__global__ void copy_cpanel(const float* __restrict__ Mm, int kb,
                            float* __restrict__ cp) {
  int idx = blockIdx.x * 256 + threadIdx.x;   // N*NB
  int r = idx >> 6, c = idx & 63;
  cp[idx] = Mm[(size_t)r * N + (size_t)kb * NB + c];
}

// ------------------------- R[j] = Binv @ M[kb][j]  (R[kb] = Binv), WMMA
__global__ __launch_bounds__(256) void compute_R(const float* __restrict__ Mm,
                                                 const float* __restrict__ binv,
                                                 int kb, float* __restrict__ R) {
  int j = blockIdx.x;
  if (j == kb) {
    for (int idx = threadIdx.x; idx < NB * NB; idx += 256) {
      int r = idx >> 6, c = idx & 63;
      R[(size_t)r * N + (size_t)j * NB + c] = binv[idx];
    }
    return;
  }
  __shared__ float sA[NB * (NB + 1)];   // Binv
  __shared__ float sB[NB * (NB + 1)];   // M[kb][j]
  tile_to_lds(sA, binv, NB);
  tile_to_lds(sB, Mm + (size_t)kb * NB * N + (size_t)j * NB, N);
  tile_sync();
  int wave = threadIdx.x >> 5, lane = threadIdx.x & 31;
  int l15 = lane & 15, kh = (lane >> 4) << 1;
  for (int t = wave; t < 16; t += 8) {          // uniform per wave
    int tM = (t >> 2) << 4, tN = (t & 3) << 4;
    v8f acc = {};
    for (int k = 0; k < NB; k += 4) {
      v2f a, b;
      a.x = sA[(tM + l15) * (NB + 1) + k + kh];
      a.y = sA[(tM + l15) * (NB + 1) + k + kh + 1];
      b.x = sB[(k + kh) * (NB + 1) + tN + l15];
      b.y = sB[(k + kh + 1) * (NB + 1) + tN + l15];
      acc = wmma4(a, b, acc);
    }
    int rbase = tM + ((lane >> 4) << 3), col = tN + l15;
#pragma unroll
    for (int r = 0; r < 8; ++r)
      R[(size_t)(rbase + r) * N + (size_t)j * NB + col] = acc[r];
  }
}

// ------ rank-NB update: M[i][j] -= Ci@R[j]; M[i][kb] = -Ci@Binv; row kb = R
__global__ __launch_bounds__(256) void gj_update(float* __restrict__ Mm,
                                                 const float* __restrict__ cp,
                                                 const float* __restrict__ binv,
                                                 const float* __restrict__ R,
                                                 int kb) {
  int j = blockIdx.x, i = blockIdx.y;
  if (i == kb) {                                 // write back scaled pivot row
    for (int idx = threadIdx.x; idx < NB * NB; idx += 256) {
      int r = idx >> 6, c = idx & 63;
      Mm[(size_t)(kb * NB + r) * N + (size_t)j * NB + c] =
          R[(size_t)r * N + (size_t)j * NB + c];
    }
    return;
  }
  __shared__ float sC[NB * (NB + 1)];   // Ci = old M[i][kb]
  __shared__ float sB[NB * (NB + 1)];   // R[j] or Binv
  bool pivcol = (j == kb);
  tile_to_lds(sC, cp + (size_t)i * NB * NB, NB);
  tile_to_lds(sB, pivcol ? binv : R + (size_t)j * NB, pivcol ? NB : N);
  tile_sync();
  int wave = threadIdx.x >> 5, lane = threadIdx.x & 31;
  int l15 = lane & 15, kh = (lane >> 4) << 1;
  for (int t = wave; t < 16; t += 8) {
    int tM = (t >> 2) << 4, tN = (t & 3) << 4;
    v8f acc = {};
    for (int k = 0; k < NB; k += 4) {
      v2f a, b;
      a.x = sC[(tM + l15) * (NB + 1) + k + kh];
      a.y = sC[(tM + l15) * (NB + 1) + k + kh + 1];
      b.x = sB[(k + kh) * (NB + 1) + tN + l15];
      b.y = sB[(k + kh + 1) * (NB + 1) + tN + l15];
      acc = wmma4(a, b, acc);
    }
    int rbase = tM + ((lane >> 4) << 3), col = tN + l15;
#pragma unroll
    for (int r = 0; r < 8; ++r) {
      size_t off = (size_t)(i * NB + rbase + r) * N + (size_t)j * NB + col;
      if (pivcol) Mm[off] = -acc[r];
      else        Mm[off] = Mm[off] - acc[r];
    }
  }
}

// ----------------------------------------- Bernoulli-probit Newton pieces
__global__ void newton_elem(const float* __restrict__ f, const float* __restrict__ y,
                            float* __restrict__ Wd, float* __restrict__ rhs) {
  int i = blockIdx.x * 256 + threadIdx.x;
  if (i >= N) return;
  float sgn = 2.0f * y[i] - 1.0f;
  float fv = f[i];
  float z = sgn * fv;
  float pdf = 0.3989422804014327f * expf(-0.5f * z * z);
  float cdf = 0.5f * erfcf(-z * 0.70710678118654752f);
  cdf = fminf(fmaxf(cdf, 1e-9f), 1.0f);
  float r = pdf / cdf;
  float w = r * r + z * r;
  Wd[i] = w;
  rhs[i] = w * fv + sgn * r;
}

__global__ void build_M(const float* __restrict__ A, const float* __restrict__ Wd,
                        float* __restrict__ Mm) {
  int idx = blockIdx.x * 256 + threadIdx.x;   // N*N = 2^22 fits int
  int r = idx >> 11, c = idx & (N - 1);
  float v = A[idx];
  if (r == c) v += Wd[r];
  Mm[idx] = v;
}

// ------------------------------------------------- f = Pinv @ rhs (GEMV)
__global__ __launch_bounds__(256) void gemv(const float* __restrict__ Mm,
                                            const float* __restrict__ rhs,
                                            float* __restrict__ fout) {
  int wave = threadIdx.x >> 5, lane = threadIdx.x & 31;
  int row = blockIdx.x * 8 + wave;
  const float* mrow = Mm + (size_t)row * N;
  float sum = 0.0f;
  for (int c = lane; c < N; c += 32) sum += mrow[c] * rhs[c];
  for (int s = 16; s > 0; s >>= 1) sum += __shfl_down(sum, s, 32);
  if (lane == 0) fout[row] = sum;
}

__global__ void finalize(const float* __restrict__ f, const float* __restrict__ Wd,
                         float* __restrict__ out_f, float* __restrict__ out_W) {
  int i = blockIdx.x * 256 + threadIdx.x;
  if (i >= N) return;
  out_f[i] = f[i];
  out_W[(size_t)i * N + i] = Wd[i];
}

// =========================================================================
extern "C" void kernel_launch(void* const* d_in, const int* in_sizes, int n_in,
                              void* d_out, int out_size, void* d_ws, size_t ws_size,
                              hipStream_t stream) {
  (void)in_sizes; (void)n_in; (void)out_size; (void)ws_size;
  const float* x    = (const float*)d_in[0];
  const float* y    = (const float*)d_in[1];
  const float* f0   = (const float*)d_in[2];
  const float* W_fe = (const float*)d_in[3];
  const float* b_fe = (const float*)d_in[4];
  const float* kvar = (const float*)d_in[5];
  const float* var  = (const float*)d_in[6];

  float* out      = (float*)d_out;
  float* out_f    = out;                                  // [N]
  float* out_Pinv = out + N;                              // [N*N] = inverse workspace
  float* out_W    = out + N + (size_t)N * N;              // [N*N]
  float* out_G    = out + N + 2 * (size_t)N * N;          // [MF*N]

  float* ws   = (float*)d_ws;
  float* A    = ws;  ws += (size_t)N * N;                 // 16 MB
  float* P    = ws;  ws += (size_t)N * MF;                // 1 MB
  float* R    = ws;  ws += (size_t)NB * N;                // 0.5 MB
  float* cp   = ws;  ws += (size_t)N * NB;                // 0.5 MB
  float* binv = ws;  ws += NB * NB;
  float* cmin = ws;  ws += MF;
  float* cmax = ws;  ws += MF;
  float* fv   = ws;  ws += N;
  float* rhs  = ws;  ws += N;
  float* Wd   = ws;  ws += N;

  float* Mm = out_Pinv;   // in-place inverse working matrix lives in d_out

  zero4<<<N * N / 1024, 256, 0, stream>>>((float4*)out_W);
  gemm_xw<<<128, 256, 0, stream>>>(x, W_fe, b_fe, P);
  col_minmax<<<MF, 256, 0, stream>>>(P, cmin, cmax);
  normalize_G<<<(N * MF) / 256, 256, 0, stream>>>(P, cmin, cmax, kvar, out_G);
  syrk_K<<<(128 * 128) / 8, 256, 0, stream>>>(P, var, Mm);

  auto invert = [&](float* Mat) {
    for (int kb = 0; kb < NBLK; ++kb) {
      inv_diag<<<1, 256, 0, stream>>>(Mat, kb, binv);
      copy_cpanel<<<(N * NB) / 256, 256, 0, stream>>>(Mat, kb, cp);
      compute_R<<<NBLK, 256, 0, stream>>>(Mat, binv, kb, R);
      dim3 g(NBLK, NBLK);
      gj_update<<<g, 256, 0, stream>>>(Mat, cp, binv, R, kb);
    }
  };

  invert(Mm);                                            // Mm = A = inv(K+varI)
  copy_mat4<<<N * N / 1024, 256, 0, stream>>>((const float4*)Mm, (float4*)A);
  copy_vec<<<N / 256, 256, 0, stream>>>(f0, fv);

  for (int it = 0; it < NSTEPS; ++it) {
    newton_elem<<<N / 256, 256, 0, stream>>>(fv, y, Wd, rhs);
    build_M<<<N * N / 256, 256, 0, stream>>>(A, Wd, Mm);
    invert(Mm);                                          // Mm = Pinv
    gemv<<<N / 8, 256, 0, stream>>>(Mm, rhs, fv);
  }
  finalize<<<N / 256, 256, 0, stream>>>(fv, Wd, out_f, out_W);
}